// Voxelization_53781580481201
// MI455X (gfx1250) — compile-verified
//
#include <hip/hip_runtime.h>
#include <math.h>

// Problem constants (from reference setup_inputs).
#define BB   16
#define CC   64
#define NN   65536          // 2^16
#define RES  32
#define R3   (RES*RES*RES)  // 32768 = 2^15
// C*R3 = 2^21

#define GLOBAL_AS __attribute__((address_space(1)))
#define LDS_AS    __attribute__((address_space(3)))

typedef int v4i __attribute__((vector_size(4 * sizeof(int))));

// ---- gfx1250 async global->LDS (ASYNCcnt) with safe fallback --------------
#if defined(__has_builtin)
#  if __has_builtin(__builtin_amdgcn_global_load_async_to_lds_b128)
#    define HAVE_ASYNC_LDS 1
#  endif
#endif

#if defined(HAVE_ASYNC_LDS)
#  define ASYNC_B128(g, l)                                                   \
      __builtin_amdgcn_global_load_async_to_lds_b128(                        \
          (GLOBAL_AS v4i*)(g), (LDS_AS v4i*)(l), 0, 0)
#  if __has_builtin(__builtin_amdgcn_s_wait_asynccnt)
#    define WAIT_ASYNC0() __builtin_amdgcn_s_wait_asynccnt(0)
#  else
#    define WAIT_ASYNC0() asm volatile("s_wait_asynccnt 0x0" ::: "memory")
#  endif
#else
#  define ASYNC_B128(g, l) (*(float4*)(l) = *(const float4*)(g))
#  define WAIT_ASYNC0() ((void)0)
#endif

// Workspace layout (floats): [0, B*R3) = counts ; [B*R3, B*R3+B*4) = stats
// stats[b] = { sum_x, sum_y, sum_z, max_sq_norm }  (accumulated atomically)

// ---------------------------------------------------------------------------
// Kernel 1: zero accumulators (d_out voxel region) + counts + stats.
// PLAIN stores: the zeroed lines must stay resident in L2 for the atomics.
// ---------------------------------------------------------------------------
__global__ void vox_zero_kernel(float* __restrict__ sums, float* __restrict__ ws) {
    const size_t i = (size_t)blockIdx.x * blockDim.x + threadIdx.x;
    const size_t n_sums = (size_t)BB * CC * R3;
    const size_t n_ws   = (size_t)BB * R3 + (size_t)BB * 4;
    if (i < n_sums) {
        sums[i] = 0.0f;                     // RT policy -> stays hot in L2
    } else if (i < n_sums + n_ws) {
        ws[i - n_sums] = 0.0f;
    }
}

// ---------------------------------------------------------------------------
// Kernel 2: partial coordinate sums.  Grid = B*64 blocks x 256 threads;
// each block stages a 3 x 1024-point slab into LDS via async b128, reduces,
// and merges with 3 global float atomics.
// ---------------------------------------------------------------------------
__global__ void vox_stats_sum_kernel(const float* __restrict__ coords,
                                     float* __restrict__ stats) {
    __shared__ __align__(16) float buf[3][1024];
    __shared__ float red[3][256];
    const int b   = blockIdx.x >> 6;
    const int seg = blockIdx.x & 63;
    const int t   = threadIdx.x;
    const float* cb = coords + (size_t)b * 3 * NN + (size_t)seg * 1024;

    #pragma unroll
    for (int d = 0; d < 3; ++d)
        ASYNC_B128(cb + (size_t)d * NN + 4 * t, &buf[d][4 * t]);
    WAIT_ASYNC0();                          // each lane reads only its own 16B

    #pragma unroll
    for (int d = 0; d < 3; ++d)
        red[d][t] = buf[d][4*t] + buf[d][4*t+1] + buf[d][4*t+2] + buf[d][4*t+3];
    __syncthreads();
    for (int s = 128; s > 0; s >>= 1) {
        if (t < s) {
            red[0][t] += red[0][t + s];
            red[1][t] += red[1][t + s];
            red[2][t] += red[2][t + s];
        }
        __syncthreads();
    }
    if (t == 0) {
        atomicAdd(&stats[b * 4 + 0], red[0][0]);
        atomicAdd(&stats[b * 4 + 1], red[1][0]);
        atomicAdd(&stats[b * 4 + 2], red[2][0]);
    }
}

// ---------------------------------------------------------------------------
// Kernel 3: partial max of squared norm about the mean.  Same shape as K2.
// Merge with integer atomicMax (valid: squared norms >= 0).
// ---------------------------------------------------------------------------
__global__ void vox_stats_max_kernel(const float* __restrict__ coords,
                                     float* __restrict__ stats) {
    __shared__ __align__(16) float buf[3][1024];
    __shared__ float red[256];
    const int b   = blockIdx.x >> 6;
    const int seg = blockIdx.x & 63;
    const int t   = threadIdx.x;
    const float* cb = coords + (size_t)b * 3 * NN + (size_t)seg * 1024;

    const float mx = stats[b * 4 + 0] * (1.0f / NN);
    const float my = stats[b * 4 + 1] * (1.0f / NN);
    const float mz = stats[b * 4 + 2] * (1.0f / NN);

    #pragma unroll
    for (int d = 0; d < 3; ++d)
        ASYNC_B128(cb + (size_t)d * NN + 4 * t, &buf[d][4 * t]);
    WAIT_ASYNC0();

    float mm = 0.0f;
    #pragma unroll
    for (int k = 0; k < 4; ++k) {
        const float x = buf[0][4 * t + k] - mx;
        const float y = buf[1][4 * t + k] - my;
        const float z = buf[2][4 * t + k] - mz;
        mm = fmaxf(mm, x * x + y * y + z * z);
    }
    red[t] = mm;
    __syncthreads();
    for (int s = 128; s > 0; s >>= 1) {
        if (t < s) red[t] = fmaxf(red[t], red[t + s]);
        __syncthreads();
    }
    if (t == 0)
        atomicMax((int*)&stats[b * 4 + 3], __float_as_int(red[0]));
}

// ---------------------------------------------------------------------------
// Kernel 4: per-point scatter.  Coords staged per block via async b128.
// 64 global_atomic_add_f32 into the L2-resident grid + 1 count atomic.
// Feature stream is non-temporal (read once) + prefetched (256KB strides).
// ---------------------------------------------------------------------------
__global__ void vox_scatter_kernel(const float* __restrict__ feat,
                                   const float* __restrict__ coords,
                                   const float* __restrict__ stats,
                                   float* __restrict__ sums,
                                   float* __restrict__ nc_out,
                                   float* __restrict__ cnt) {
    __shared__ __align__(16) float cbuf[3][256];
    const int blk = blockIdx.x;                 // 4096 blocks, 256/batch
    const int b   = blk >> 8;
    const int n0  = (blk & 255) << 8;
    const int t   = threadIdx.x;
    const float* cb = coords + (size_t)b * 3 * NN + n0;

    if (t < 192) {                              // waves 0..5 stage 3x1KB slabs
        const int d = t >> 6;
        const int o = (t & 63) << 2;
        ASYNC_B128(cb + (size_t)d * NN + o, &cbuf[d][o]);
    }
    WAIT_ASYNC0();
    __syncthreads();                            // cross-wave LDS consumption

    // Uniform per block -> scalar loads / scalar math.
    const float mx  = stats[b * 4 + 0] * (1.0f / NN);
    const float my  = stats[b * 4 + 1] * (1.0f / NN);
    const float mz  = stats[b * 4 + 2] * (1.0f / NN);
    const float inv = 1.0f / (2.0f * sqrtf(stats[b * 4 + 3]));   // EPS == 0

    const int n = n0 + t;
    float x = (cbuf[0][t] - mx) * inv + 0.5f;
    float y = (cbuf[1][t] - my) * inv + 0.5f;
    float z = (cbuf[2][t] - mz) * inv + 0.5f;

    x = fminf(fmaxf(x * (float)RES, 0.0f), (float)(RES - 1));
    y = fminf(fmaxf(y * (float)RES, 0.0f), (float)(RES - 1));
    z = fminf(fmaxf(z * (float)RES, 0.0f), (float)(RES - 1));

    // nc output (second tuple element), streaming writes.
    float* nb = nc_out + (size_t)b * 3 * NN;
    __builtin_nontemporal_store(x, &nb[n]);
    __builtin_nontemporal_store(y, &nb[NN + n]);
    __builtin_nontemporal_store(z, &nb[2 * NN + n]);

    // jnp.round == round-half-to-even.
    const int vx = __float2int_rn(x);
    const int vy = __float2int_rn(y);
    const int vz = __float2int_rn(z);
    const int idx = (vx * RES + vy) * RES + vz;

    const float* fb = feat + (size_t)b * CC * NN + n;
    float*       sb = sums + (size_t)b * CC * R3 + idx;

    #pragma unroll 8
    for (int c = 0; c < CC; ++c) {
        if (c + 8 < CC)
            __builtin_prefetch(fb + (size_t)(c + 8) * NN, 0, 0);   // global_prefetch_b8
        const float v = __builtin_nontemporal_load(fb + (size_t)c * NN);
        atomicAdd(sb + (size_t)c * R3, v);                         // L2-resident RMW
    }
    atomicAdd(cnt + (size_t)b * R3 + idx, 1.0f);
}

// ---------------------------------------------------------------------------
// Kernel 5: sums /= max(cnt, 1).  In-place; final write is streaming.
// ---------------------------------------------------------------------------
__global__ void vox_finalize_kernel(float* __restrict__ sums,
                                    const float* __restrict__ cnt) {
    const size_t e = (size_t)blockIdx.x * blockDim.x + threadIdx.x;
    if (e >= (size_t)BB * CC * R3) return;
    const size_t v = e & (size_t)(R3 - 1);
    const size_t b = e >> 21;                   // C*R3 = 2^21
    const float c = cnt[b * R3 + v];
    const float out = sums[e] / fmaxf(c, 1.0f);
    __builtin_nontemporal_store(out, &sums[e]);
}

// ---------------------------------------------------------------------------
extern "C" void kernel_launch(void* const* d_in, const int* in_sizes, int n_in,
                              void* d_out, int out_size, void* d_ws, size_t ws_size,
                              hipStream_t stream) {
    (void)in_sizes; (void)n_in; (void)out_size; (void)ws_size;

    const float* feat   = (const float*)d_in[0];   // [B, C, N]
    const float* coords = (const float*)d_in[1];   // [B, 3, N]
    // d_in[2] = resolution (scalar 32), baked into constants.

    float* sums   = (float*)d_out;                          // [B, C, R3]
    float* nc_out = (float*)d_out + (size_t)BB * CC * R3;   // [B, 3, N]

    float* ws    = (float*)d_ws;
    float* cnt   = ws;                                      // [B, R3]
    float* stats = ws + (size_t)BB * R3;                    // [B, 4]

    {   // 1) zero accumulators + counts + stats
        const size_t total = (size_t)BB * CC * R3 + (size_t)BB * R3 + BB * 4;
        vox_zero_kernel<<<(int)((total + 255) / 256), 256, 0, stream>>>(sums, ws);
    }
    // 2) partial sums -> mean accumulators
    vox_stats_sum_kernel<<<BB * 64, 256, 0, stream>>>(coords, stats);
    // 3) partial max squared-norm
    vox_stats_max_kernel<<<BB * 64, 256, 0, stream>>>(coords, stats);
    // 4) scatter-add (one thread per point)
    vox_scatter_kernel<<<(BB * NN) / 256, 256, 0, stream>>>(feat, coords, stats,
                                                            sums, nc_out, cnt);
    // 5) divide by counts
    vox_finalize_kernel<<<(int)(((size_t)BB * CC * R3) / 256), 256, 0, stream>>>(sums, cnt);
}